// CascadingSinkCacheCompile_26980984553671
// MI455X (gfx1250) — compile-verified
//
#include <hip/hip_runtime.h>
#include <cstdint>
#include <cstddef>

// ---------------- problem constants ----------------
#define Bn 2
#define Hn 32
#define Dn 128
#define Tn 2048
#define Wn 512
#define Sn 4096
#define NSINKn 4
#define Cn (Sn / Wn) // 8

// ---------------- output layout (flat float elements) ----------------
#define OFF_SINK_K   ((size_t)0)
#define OFF_SINK_V   ((size_t)(Bn * Hn * NSINKn * Dn))             // 32768
#define OFF_SINK_POS ((size_t)(2 * Bn * Hn * NSINKn * Dn))         // 65536
#define OFF_SINK_MSK (OFF_SINK_POS + NSINKn)                       // 65540
#define OFF_KC       (OFF_SINK_MSK + NSINKn)                       // 65544
#define KV_ELEMS     ((size_t)Bn * Hn * Sn * Dn)                   // 33554432
#define OFF_VC       (OFF_KC + KV_ELEMS)
#define OFF_POS      (OFF_VC + KV_ELEMS)
#define OFF_MSK      (OFF_POS + Sn)

// ---------------- compile-time cascade simulation ----------------
// The cache control flow is data-independent (score cache is identically zero,
// so overwrite-compare always succeeds). Simulate the whole T-token stream at
// compile time and embed the slot->token map + positions + occupancy.
struct Sim {
    short tok[Sn];         // original token index last written to slot, -1 if never
    short pos[Sn];         // pos_out[s]
    unsigned char occ[Sn]; // occupancy (mask = occ ? 0 : -FLT_MAX)
};

constexpr Sim make_sim() {
    Sim r{};
    int starts[Cn] = {};
    int stored[Cn] = {};
    for (int s = 0; s < Sn; ++s) r.tok[s] = (short)-1;

    for (int t = NSINKn; t < Tn; ++t) {
        int m = t - NSINKn;
        int cur = t;
        bool active = true;
        for (int i = 0; i < Cn && active; ++i) {
            const int l = i * Wn;
            const int st = starts[i];
            const int ntok = stored[i];
            const bool full = (ntok == Wn);
            const bool dc = ((m & ((1 << i) - 1)) == 0);
            const bool append_f = (dc && !full) || (!dc && ntok == 0);
            const bool evict_f  = dc && full;
            const bool ow_f     = !dc && (ntok > 0);  // score compare always true
            const int widx = append_f ? (l + (st + ntok) % Wn) : (l + st);
            const int old_tok = r.tok[l + st];        // read before write
            r.tok[widx] = (short)cur;                 // when active a write always happens
            if (append_f) stored[i] = ntok + 1;
            if (evict_f || ow_f) starts[i] = (st + 1) % Wn;
            if (evict_f) { cur = old_tok; active = true; }
            else         { active = false; }
        }
    }
    // positions / occupancy, matching the reference epilogue
    int prefix[Cn] = {};
    int acc = 0;
    for (int i = 0; i < Cn; ++i) { acc += stored[i]; prefix[i] = acc; }
    const int total = acc;
    for (int s = 0; s < Sn; ++s) {
        const int casc = s / Wn;
        const int local = s % Wn;
        const int rank = ((local - starts[casc]) % Wn + Wn) % Wn;
        const bool o = rank < stored[casc];
        r.occ[s] = o ? 1u : 0u;
        const int p = o ? (total - prefix[casc] + rank) : 0;
        r.pos[s] = (short)(p + NSINKn);
    }
    return r;
}

__device__ __constant__ Sim g_sim = make_sim();

// ---------------- CDNA5 async global<->LDS copy path ----------------
typedef int v4i __attribute__((vector_size(16)));
#define AS1 __attribute__((address_space(1)))
#define AS3 __attribute__((address_space(3)))

#if defined(__has_builtin)
#if __has_builtin(__builtin_amdgcn_global_load_async_to_lds_b128) && \
    __has_builtin(__builtin_amdgcn_global_store_async_from_lds_b128)
#define USE_ASYNC_BUILTIN 1
#endif
#endif
#ifndef USE_ASYNC_BUILTIN
#define USE_ASYNC_BUILTIN 0
#endif

__device__ __forceinline__ void async_g2l_b128(const float* g, float* l) {
#if USE_ASYNC_BUILTIN
    __builtin_amdgcn_global_load_async_to_lds_b128(
        (AS1 v4i*)g, (AS3 v4i*)l, 0, 0);
#else
    asm volatile("global_load_async_to_lds_b128 %0, %1, off"
                 :
                 : "v"((AS3 v4i*)l),
                   "v"((AS1 v4i*)g)
                 : "memory");
#endif
}

__device__ __forceinline__ void async_l2g_b128(float* g, const float* l) {
#if USE_ASYNC_BUILTIN
    __builtin_amdgcn_global_store_async_from_lds_b128(
        (AS1 v4i*)g, (AS3 v4i*)l, 0, 0);
#else
    asm volatile("global_store_async_from_lds_b128 %0, %1, off"
                 :
                 : "v"((AS1 v4i*)g),
                   "v"((AS3 v4i*)l)
                 : "memory");
#endif
}

__device__ __forceinline__ void wait_async0() {
#if defined(__has_builtin) && __has_builtin(__builtin_amdgcn_s_wait_asynccnt)
    __builtin_amdgcn_s_wait_asynccnt(0);
#else
    asm volatile("s_wait_asynccnt 0x0" ::: "memory");
#endif
}

// ---------------- kernels ----------------
#define WAVES_PER_BLOCK 8

// One wave32 per (b*h, s) row: copies the 512B K row and 512B V row.
// Occupied rows stream through LDS via the async data mover; empty rows store 0.
__global__ __launch_bounds__(256) void gather_kv(const float* __restrict__ key,
                                                 const float* __restrict__ val,
                                                 float* __restrict__ out) {
    __shared__ float stage[WAVES_PER_BLOCK * 2 * Dn]; // 8 KB
    const int lane  = threadIdx.x & 31;
    const int wslot = threadIdx.x >> 5;
    const long wave = (long)blockIdx.x * WAVES_PER_BLOCK + wslot;
    if (wave >= (long)Bn * Hn * Sn) return;

    const int s  = (int)(wave & (Sn - 1));
    const int bh = (int)(wave >> 12); // Sn == 2^12
    const int tok = g_sim.tok[s];

    float* dk = out + OFF_KC + ((size_t)bh * Sn + s) * Dn;
    float* dv = out + OFF_VC + ((size_t)bh * Sn + s) * Dn;

    if (tok < 0) {
        const float4 z = make_float4(0.f, 0.f, 0.f, 0.f);
        ((float4*)dk)[lane] = z;
        ((float4*)dv)[lane] = z;
        return;
    }

    const float* sk = key + ((size_t)bh * Tn + tok) * Dn;
    const float* sv = val + ((size_t)bh * Tn + tok) * Dn;
    float* lk = &stage[wslot * 2 * Dn];
    float* lv = lk + Dn;

    async_g2l_b128(sk + lane * 4, lk + lane * 4);
    async_g2l_b128(sv + lane * 4, lv + lane * 4);
    wait_async0();                       // loads landed in LDS
    async_l2g_b128(dk + lane * 4, lk + lane * 4);
    async_l2g_b128(dv + lane * 4, lv + lane * 4);
    // s_endpgm performs an implicit s_wait_idle, draining ASYNCcnt stores.
}

// sink_k / sink_v / sink_pos / sink_mask / pos_out / mask
__global__ __launch_bounds__(256) void fill_meta(const float* __restrict__ key,
                                                 const float* __restrict__ val,
                                                 float* __restrict__ out) {
    const int i = blockIdx.x * blockDim.x + threadIdx.x;
    if (i < Bn * Hn * NSINKn * Dn) {
        const int d  = i & (Dn - 1);
        const int j  = (i >> 7) & (NSINKn - 1);
        const int bh = i >> 9;
        const size_t src = ((size_t)bh * Tn + j) * Dn + d;
        out[OFF_SINK_K + i] = key[src];
        out[OFF_SINK_V + i] = val[src];
    }
    if (i < NSINKn) {
        out[OFF_SINK_POS + i] = (float)i; // stored_sinks == NSINK, all occupied
        out[OFF_SINK_MSK + i] = 0.0f;
    }
    if (i < Sn) {
        out[OFF_POS + i] = (float)g_sim.pos[i];
        out[OFF_MSK + i] = g_sim.occ[i] ? 0.0f : -__FLT_MAX__;
    }
}

// ---------------- launch ----------------
extern "C" void kernel_launch(void* const* d_in, const int* in_sizes, int n_in,
                              void* d_out, int out_size, void* d_ws, size_t ws_size,
                              hipStream_t stream) {
    (void)in_sizes; (void)n_in; (void)out_size; (void)d_ws; (void)ws_size;
    const float* key = (const float*)d_in[0];
    const float* val = (const float*)d_in[1];
    float* out = (float*)d_out;

    // metadata + sinks (32768 threads covers every range used inside)
    fill_meta<<<(Bn * Hn * NSINKn * Dn + 255) / 256, 256, 0, stream>>>(key, val, out);

    // main gather: one wave per cache row
    const long nwaves = (long)Bn * Hn * Sn; // 262144
    const int blocks = (int)((nwaves + WAVES_PER_BLOCK - 1) / WAVES_PER_BLOCK);
    gather_kv<<<blocks, 256, 0, stream>>>(key, val, out);
}